// OptimizedSimpleGAT_28441273434161
// MI455X (gfx1250) — compile-verified
//
#include <hip/hip_runtime.h>
#include <hip/hip_bf16.h>

#define N_NODES   50000
#define N_EDGES   1600000
#define E_TOT     1650000   // + self loops
#define N_GRAPHS  512
#define NEG_SLOPE 0.2f

typedef __attribute__((ext_vector_type(2))) float v2f;
typedef __attribute__((ext_vector_type(8))) float v8f;

#if defined(__has_builtin)
#if __has_builtin(__builtin_amdgcn_tensor_load_to_lds) && __has_builtin(__builtin_amdgcn_s_wait_tensorcnt)
#define HAVE_TDM 1
#endif
#endif
#ifndef HAVE_TDM
#define HAVE_TDM 0
#endif

#if HAVE_TDM
typedef unsigned int v4u  __attribute__((ext_vector_type(4)));
typedef int          v8i_ __attribute__((ext_vector_type(8)));
typedef int          v4i_ __attribute__((ext_vector_type(4)));
#endif

__device__ __forceinline__ float lrelu(float v) { return v > 0.f ? v : NEG_SLOPE * v; }

__device__ __forceinline__ void get_edge(const int* __restrict__ ei, int e, int& s, int& d) {
    if (e < N_EDGES) { s = ei[e]; d = ei[N_EDGES + e]; }
    else             { s = e - N_EDGES; d = s; }      // self loop
}

__device__ __forceinline__ void atomicMaxF(float* addr, float val) {
    int cur = __float_as_int(*addr);
    while (val > __int_as_float(cur)) {
        int prev = atomicCAS(reinterpret_cast<int*>(addr), cur, __float_as_int(val));
        if (prev == cur) break;
        cur = prev;
    }
}

// ---------------- layer 1: x(50000x5) @ W1l/W1r (5x64) ----------------
__global__ __launch_bounds__(256) void k_node_gemm5(
    const float* __restrict__ x, const float* __restrict__ Wl,
    const float* __restrict__ Wr, float* __restrict__ xl, float* __restrict__ xr) {
    int tid = blockIdx.x * 256 + threadIdx.x;            // 3.2M threads
    int n = tid >> 6, f = tid & 63;
    float sl = 0.f, sr = 0.f;
#pragma unroll
    for (int k = 0; k < 5; ++k) {
        float xv = x[n * 5 + k];
        sl = fmaf(xv, Wl[k * 64 + f], sl);
        sr = fmaf(xv, Wr[k * 64 + f], sr);
    }
    xl[tid] = sl; xr[tid] = sr;
}

__global__ __launch_bounds__(256) void k_init1(float* __restrict__ h1,
    float* __restrict__ m1, float* __restrict__ s1) {
    int tid = blockIdx.x * 256 + threadIdx.x;            // 3.2M threads
    h1[tid] = 0.f;
    if (tid < N_NODES * 4) { m1[tid] = -3.402823466e38f; s1[tid] = 0.f; }
}

__global__ __launch_bounds__(256) void k_edge_logits1(
    const int* __restrict__ ei, const float* __restrict__ xl,
    const float* __restrict__ xr, const float* __restrict__ att,
    float* __restrict__ e1, float* __restrict__ m1) {
    int tid = blockIdx.x * 256 + threadIdx.x;
    if (tid >= E_TOT * 4) return;
    int e = tid >> 2, h = tid & 3;
    int s, d; get_edge(ei, e, s, d);
    const float4* pl = reinterpret_cast<const float4*>(xl + s * 64 + h * 16);
    const float4* pr = reinterpret_cast<const float4*>(xr + d * 64 + h * 16);
    const float4* pa = reinterpret_cast<const float4*>(att + h * 16);
    float acc = 0.f;
#pragma unroll
    for (int i = 0; i < 4; ++i) {
        float4 a = pl[i], b = pr[i], w = pa[i];
        acc = fmaf(lrelu(a.x + b.x), w.x, acc);
        acc = fmaf(lrelu(a.y + b.y), w.y, acc);
        acc = fmaf(lrelu(a.z + b.z), w.z, acc);
        acc = fmaf(lrelu(a.w + b.w), w.w, acc);
    }
    e1[tid] = acc;
    atomicMaxF(&m1[d * 4 + h], acc);
}

__global__ __launch_bounds__(256) void k_edge_exp1(
    const int* __restrict__ ei, float* __restrict__ e1,
    const float* __restrict__ m1, float* __restrict__ s1) {
    int tid = blockIdx.x * 256 + threadIdx.x;
    if (tid >= E_TOT * 4) return;
    int e = tid >> 2, h = tid & 3;
    int s, d; get_edge(ei, e, s, d);
    float p = __expf(e1[tid] - m1[d * 4 + h]);
    e1[tid] = p;
    atomicAdd(&s1[d * 4 + h], p);
}

__global__ __launch_bounds__(256) void k_aggregate1(
    const int* __restrict__ ei, const float* __restrict__ e1,
    const float* __restrict__ s1, const float* __restrict__ xl,
    float* __restrict__ h1) {
    int tid = blockIdx.x * 256 + threadIdx.x;
    if (tid >= E_TOT * 4) return;
    int e = tid >> 2, h = tid & 3;
    int s, d; get_edge(ei, e, s, d);
    float alpha = e1[tid] / (s1[d * 4 + h] + 1e-16f);
    const float4* pl = reinterpret_cast<const float4*>(xl + s * 64 + h * 16);
    float* dp = h1 + d * 64 + h * 16;
#pragma unroll
    for (int i = 0; i < 4; ++i) {
        float4 v = pl[i];
        atomicAdd(dp + 4 * i + 0, v.x * alpha);
        atomicAdd(dp + 4 * i + 1, v.y * alpha);
        atomicAdd(dp + 4 * i + 2, v.z * alpha);
        atomicAdd(dp + 4 * i + 3, v.w * alpha);
    }
}

__global__ __launch_bounds__(256) void k_finalize1(float* __restrict__ h1,
    const float* __restrict__ b1) {
    int tid = blockIdx.x * 256 + threadIdx.x;            // 3.2M
    float v = h1[tid] + b1[tid & 63];
    h1[tid] = v > 0.f ? v : 0.f;
}

// ---------------- layer 2 GEMM: h1(50000x64) @ W2l/W2r (64x64) via WMMA ----------------
// One block = one 16-row tile of h1 staged in LDS (via TDM tensor_load_to_lds when
// available); 8 waves: waves 0-3 -> W2l n-tiles, waves 4-7 -> W2r n-tiles.
// Each wave: 16 x V_WMMA_F32_16X16X4_F32 (K=64).
__global__ __launch_bounds__(256) void k_gemm64_wmma(
    const float* __restrict__ h, const float* __restrict__ Wl,
    const float* __restrict__ Wr, float* __restrict__ xl, float* __restrict__ xr) {
    __shared__ float ldsA[16 * 64];
    const int m0 = blockIdx.x * 16;
    const int t = threadIdx.x;

#if HAVE_TDM
    if (t < 32) {   // wave 0 issues one TDM DMA for the 16x64 f32 A tile
        unsigned long long ga = (unsigned long long)(const void*)(h + m0 * 64);
        unsigned int ldsb = (unsigned int)(unsigned long long)(void*)ldsA;
        v4u g0;
        g0[0] = 1u;                                        // count=1, user descriptor
        g0[1] = ldsb;                                      // lds_addr (bytes)
        g0[2] = (unsigned int)(ga & 0xFFFFFFFFull);        // global_addr[31:0]
        g0[3] = (unsigned int)((ga >> 32) & 0x01FFFFFFull) // global_addr[56:32]
              | (2u << 30);                                // type=2 ("image")
        v8i_ g1;
        g1[0] = (int)(2u << 16);                 // data_size=4B; no multicast/pad/iter
        g1[1] = (int)(64u << 16);                // tensor_dim0 = 64 (lo16)
        g1[2] = (int)(50000u << 16);             // tensor_dim0 hi=0 | tensor_dim1 lo
        g1[3] = (int)((50000u >> 16) | (64u << 16)); // tensor_dim1 hi | tile_dim0=64
        g1[4] = 16;                              // tile_dim1=16, tile_dim2=0 (2-D)
        g1[5] = 64;                              // tensor_dim0_stride = 64
        g1[6] = 0;                               // stride0 hi | stride1 lo (unused, 2-D)
        g1[7] = 0;
        v4i_ z4 = {0, 0, 0, 0};                  // groups 2/3 disabled (<=2-D tensor)
        v8i_ z8 = {0, 0, 0, 0, 0, 0, 0, 0};
        // 6-arg form (amdgpu-toolchain / clang-23 therock headers)
        __builtin_amdgcn_tensor_load_to_lds(g0, g1, z4, z4, z8, 0);
        __builtin_amdgcn_s_wait_tensorcnt(0);
    }
    __syncthreads();
#else
    // fallback: stage A tile with vector loads, 256 threads x float4
    reinterpret_cast<float4*>(ldsA)[t] =
        reinterpret_cast<const float4*>(h + m0 * 64)[t];
    __syncthreads();
#endif

    const int wave = t >> 5;
    const int lane = t & 31;
    const int half = lane >> 4;      // 0: K pair {0,1}, 1: K pair {2,3}
    const int l16  = lane & 15;
    const int n0   = (wave & 3) * 16;
    const float* W  = (wave < 4) ? Wl : Wr;
    float* out      = (wave < 4) ? xl : xr;
    __builtin_prefetch(W + n0, 0, 0);            // global_prefetch_b8 of B panel

    v8f acc = {};
#pragma unroll
    for (int k = 0; k < 64; k += 4) {
        v2f a, b;
        int ao = l16 * 64 + k + 2 * half;          // A: 16x4, row=l16, Ks split by half
        a.x = ldsA[ao];
        a.y = ldsA[ao + 1];
        int bo = (k + 2 * half) * 64 + n0 + l16;   // B: 4x16, row=K, col across lanes
        b.x = W[bo];
        b.y = W[bo + 64];
        acc = __builtin_amdgcn_wmma_f32_16x16x4_f32(
            false, a, false, b, (short)0, acc, false, false);
    }
#pragma unroll
    for (int r = 0; r < 8; ++r) {                  // C/D: VGPR r = row r (+8 upper half)
        int row = m0 + half * 8 + r;
        out[row * 64 + n0 + l16] = acc[r];
    }
}

// ---------------- layer 2 attention / pool / predict ----------------
__global__ __launch_bounds__(256) void k_init2(float* __restrict__ h2,
    float* __restrict__ m2, float* __restrict__ s2,
    float* __restrict__ pool, float* __restrict__ cnt) {
    int tid = blockIdx.x * 256 + threadIdx.x;            // 3.2M
    h2[tid] = 0.f;
    if (tid < N_NODES) { m2[tid] = -3.402823466e38f; s2[tid] = 0.f; }
    if (tid < N_GRAPHS * 64) pool[tid] = 0.f;
    if (tid < N_GRAPHS) cnt[tid] = 0.f;
}

__global__ __launch_bounds__(256) void k_edge_logits2(
    const int* __restrict__ ei, const float* __restrict__ xl,
    const float* __restrict__ xr, const float* __restrict__ att,
    float* __restrict__ e2, float* __restrict__ m2) {
    int e = blockIdx.x * 256 + threadIdx.x;
    if (e >= E_TOT) return;
    int s, d; get_edge(ei, e, s, d);
    const float4* pl = reinterpret_cast<const float4*>(xl + s * 64);
    const float4* pr = reinterpret_cast<const float4*>(xr + d * 64);
    const float4* pa = reinterpret_cast<const float4*>(att);
    float acc = 0.f;
#pragma unroll
    for (int i = 0; i < 16; ++i) {
        float4 a = pl[i], b = pr[i], w = pa[i];
        acc = fmaf(lrelu(a.x + b.x), w.x, acc);
        acc = fmaf(lrelu(a.y + b.y), w.y, acc);
        acc = fmaf(lrelu(a.z + b.z), w.z, acc);
        acc = fmaf(lrelu(a.w + b.w), w.w, acc);
    }
    e2[e] = acc;
    atomicMaxF(&m2[d], acc);
}

__global__ __launch_bounds__(256) void k_edge_exp2(
    const int* __restrict__ ei, float* __restrict__ e2,
    const float* __restrict__ m2, float* __restrict__ s2) {
    int e = blockIdx.x * 256 + threadIdx.x;
    if (e >= E_TOT) return;
    int s, d; get_edge(ei, e, s, d);
    float p = __expf(e2[e] - m2[d]);
    e2[e] = p;
    atomicAdd(&s2[d], p);
}

__global__ __launch_bounds__(256) void k_aggregate2(
    const int* __restrict__ ei, const float* __restrict__ e2,
    const float* __restrict__ s2, const float* __restrict__ xl,
    float* __restrict__ h2) {
    int tid = blockIdx.x * 256 + threadIdx.x;            // E_TOT*4: edge x 16-wide chunk
    if (tid >= E_TOT * 4) return;
    int e = tid >> 2, q = tid & 3;
    int s, d; get_edge(ei, e, s, d);
    float alpha = e2[e] / (s2[d] + 1e-16f);
    const float4* pl = reinterpret_cast<const float4*>(xl + s * 64 + q * 16);
    float* dp = h2 + d * 64 + q * 16;
#pragma unroll
    for (int i = 0; i < 4; ++i) {
        float4 v = pl[i];
        atomicAdd(dp + 4 * i + 0, v.x * alpha);
        atomicAdd(dp + 4 * i + 1, v.y * alpha);
        atomicAdd(dp + 4 * i + 2, v.z * alpha);
        atomicAdd(dp + 4 * i + 3, v.w * alpha);
    }
}

__global__ __launch_bounds__(256) void k_finalize2_pool(
    const float* __restrict__ h2, const float* __restrict__ b2,
    const int* __restrict__ batch, float* __restrict__ pool,
    float* __restrict__ cnt) {
    int tid = blockIdx.x * 256 + threadIdx.x;            // 3.2M
    int n = tid >> 6, f = tid & 63;
    float v = h2[tid] + b2[f];
    v = v > 0.f ? v : 0.f;
    int g = batch[n];
    atomicAdd(&pool[g * 64 + f], v);
    if (f == 0) atomicAdd(&cnt[g], 1.0f);
}

__global__ __launch_bounds__(256) void k_predict(
    const float* __restrict__ pool, const float* __restrict__ cnt,
    const float* __restrict__ Wp, const float* __restrict__ bp,
    float* __restrict__ out) {
    int g = blockIdx.x * 256 + threadIdx.x;
    if (g >= N_GRAPHS) return;
    float c = fmaxf(cnt[g], 1.0f);
    float acc = bp[0];
#pragma unroll
    for (int d = 0; d < 64; ++d) acc = fmaf(pool[g * 64 + d] / c, Wp[d], acc);
    out[g] = acc;
}

extern "C" void kernel_launch(void* const* d_in, const int* in_sizes, int n_in,
                              void* d_out, int out_size, void* d_ws, size_t ws_size,
                              hipStream_t stream) {
    const float* x    = (const float*)d_in[0];
    const int*   ei   = (const int*)d_in[1];     // [2, 1.6M] (src row, dst row)
    const int*   bat  = (const int*)d_in[2];
    const float* W1l  = (const float*)d_in[3];
    const float* W1r  = (const float*)d_in[4];
    const float* att1 = (const float*)d_in[5];
    const float* b1   = (const float*)d_in[6];
    const float* W2l  = (const float*)d_in[7];
    const float* W2r  = (const float*)d_in[8];
    const float* att2 = (const float*)d_in[9];
    const float* b2   = (const float*)d_in[10];
    const float* Wp   = (const float*)d_in[11];
    const float* bp   = (const float*)d_in[12];
    float* out = (float*)d_out;

    // workspace layout (floats); layer-2 buffers reuse dead layer-1 regions
    float* F   = (float*)d_ws;
    float* xl1 = F;                   // 3.2M  (reused as xl2 after aggregate1)
    float* xr1 = F + 3200000;         // 3.2M  (reused as xr2)
    float* e1  = F + 6400000;         // 6.6M
    float* e2  = e1;                  // 1.65M (e1 dead after aggregate1)
    float* h2a = F + 8100000;         // 3.2M  (inside dead e1 region, disjoint from e2)
    float* m1  = F + 13000000;        // 200k
    float* s1  = F + 13200000;        // 200k
    float* m2  = m1;                  // 50k  (reuse)
    float* s2  = s1;                  // 50k
    float* h1  = F + 13400000;        // 3.2M
    float* pool= F + 16600000;        // 32768
    float* cnt = F + 16632768;        // 512
    float* xl2 = xl1;
    float* xr2 = xr1;
    (void)ws_size; (void)n_in; (void)in_sizes; (void)out_size;

    const int B = 256;
    const int gNode64 = (N_NODES * 64) / B;               // 12500
    const int gEdge4  = (E_TOT * 4 + B - 1) / B;          // 25782
    const int gEdge   = (E_TOT + B - 1) / B;              // 6446

    // layer 1
    k_node_gemm5<<<gNode64, B, 0, stream>>>(x, W1l, W1r, xl1, xr1);
    k_init1<<<gNode64, B, 0, stream>>>(h1, m1, s1);
    k_edge_logits1<<<gEdge4, B, 0, stream>>>(ei, xl1, xr1, att1, e1, m1);
    k_edge_exp1<<<gEdge4, B, 0, stream>>>(ei, e1, m1, s1);
    k_aggregate1<<<gEdge4, B, 0, stream>>>(ei, e1, s1, xl1, h1);
    k_finalize1<<<gNode64, B, 0, stream>>>(h1, b1);

    // layer 2 node transform (WMMA + TDM staging)
    k_gemm64_wmma<<<N_NODES / 16, B, 0, stream>>>(h1, W2l, W2r, xl2, xr2);

    // layer 2 attention + pool + predict
    k_init2<<<gNode64, B, 0, stream>>>(h2a, m2, s2, pool, cnt);
    k_edge_logits2<<<gEdge, B, 0, stream>>>(ei, xl2, xr2, att2, e2, m2);
    k_edge_exp2<<<gEdge, B, 0, stream>>>(ei, e2, m2, s2);
    k_aggregate2<<<gEdge4, B, 0, stream>>>(ei, e2, s2, xl2, h2a);
    k_finalize2_pool<<<gNode64, B, 0, stream>>>(h2a, b2, bat, pool, cnt);
    k_predict<<<(N_GRAPHS + B - 1) / B, B, 0, stream>>>(pool, cnt, Wp, bp, out);
}